// HighOrderActivation_33354716021638
// MI455X (gfx1250) — compile-verified
//
#include <hip/hip_runtime.h>

typedef __attribute__((ext_vector_type(2))) float v2f;
typedef __attribute__((ext_vector_type(4))) float v4f;
typedef __attribute__((ext_vector_type(8))) float v8f;

namespace {
constexpr int Bsz  = 2048;
constexpr int Gsz  = 512;
constexpr int Ksz  = 4;
constexpr int Dsz  = 32;
constexpr int Prow = 16; // 2^K table rows

__device__ __forceinline__ void cswap(float& a, float& b, int& ia, int& ib) {
  bool sw = b < a;
  float tf = sw ? b : a;
  float tg = sw ? a : b;
  a = tf; b = tg;
  int ti = sw ? ib : ia;
  int tj = sw ? ia : ib;
  ia = ti; ib = tj;
}
} // namespace

// One wave computes out[b_base..b_base+15, g, 0..31] as
//   W(16x16, sparse scatter of coef at idx) x params[g](16x32)
// via 8 chained V_WMMA_F32_16X16X4_F32 (4 K-chunks x 2 N-tiles).
__global__ __launch_bounds__(256) void hoact_wmma(const float* __restrict__ X,
                                                  const float* __restrict__ P,
                                                  float* __restrict__ Out) {
  const int lane   = (int)(threadIdx.x & 31u);
  const int wave   = (int)(threadIdx.x >> 5);
  const int g      = (int)blockIdx.x;
  const int m      = lane & 15;   // row within 16-row tile / N column
  const int half   = lane >> 4;   // K sub-column selector per ISA A layout
  const int b_base = (int)blockIdx.y * 128 + wave * 16;
  const int b      = b_base + m;

  const float* pg = P + (size_t)g * (Prow * Dsz);
  __builtin_prefetch(pg, 0, 3);               // global_prefetch_b8 of the 2KB table

  // ---- load X row (16B aligned) and sort-4 network with index tracking ----
  v4f x = *(const v4f*)(X + ((size_t)b * Gsz + g) * Ksz);
  float v0 = x.x, v1 = x.y, v2 = x.z, v3 = x.w;
  int i0 = 0, i1 = 1, i2 = 2, i3 = 3;
  cswap(v0, v1, i0, i1);
  cswap(v2, v3, i2, i3);
  cswap(v0, v2, i0, i2);
  cswap(v1, v3, i1, i3);
  cswap(v1, v2, i1, i2);

  // first-difference coefficients and subset-sum indices (j0 == 15 always)
  const float c0 = v0, c1 = v1 - v0, c2 = v2 - v1, c3 = v3 - v2;
  const int j3 = 1 << i3;
  const int j2 = j3 + (1 << i2);
  const int j1 = j2 + (1 << i1);
  const int j0 = 15;

  // ---- build A (sparse W row) chunks + load B (params) chunks ----
  v2f a[4], b0[4], b1[4];
#pragma unroll
  for (int c = 0; c < 4; ++c) {
    const int ja = 4 * c + 2 * half;   // K column for .x
    const int jb = ja + 1;             // K column for .y
    float w0 = 0.f, w1 = 0.f;
    w0 = (ja == j0) ? c0 : w0;
    w0 = (ja == j1) ? c1 : w0;
    w0 = (ja == j2) ? c2 : w0;
    w0 = (ja == j3) ? c3 : w0;
    w1 = (jb == j0) ? c0 : w1;
    w1 = (jb == j1) ? c1 : w1;
    w1 = (jb == j2) ? c2 : w1;
    w1 = (jb == j3) ? c3 : w1;
    a[c].x = w0;
    a[c].y = w1;
    // B layout (4x16 f32): lane n holds column n; .x/.y are K rows ja/jb
    b0[c].x = pg[ja * Dsz + m];
    b0[c].y = pg[jb * Dsz + m];
    b1[c].x = pg[ja * Dsz + 16 + m];
    b1[c].y = pg[jb * Dsz + 16 + m];
  }

  // ---- 8x V_WMMA_F32_16X16X4_F32, K accumulated across 4 chunks ----
  v8f acc0 = {};
  v8f acc1 = {};
#pragma unroll
  for (int c = 0; c < 4; ++c) {
    acc0 = __builtin_amdgcn_wmma_f32_16x16x4_f32(false, a[c], false, b0[c],
                                                 (short)0, acc0, false, false);
    acc1 = __builtin_amdgcn_wmma_f32_16x16x4_f32(false, a[c], false, b1[c],
                                                 (short)0, acc1, false, false);
  }

  // ---- store D: VGPR r -> M = r + 8*half, lane -> N = m ----
#pragma unroll
  for (int r = 0; r < 8; ++r) {
    const int row = r + 8 * half;
    const size_t base = ((size_t)(b_base + row) * Gsz + g) * Dsz;
    Out[base + m]      = acc0[r];
    Out[base + 16 + m] = acc1[r];
  }
}

extern "C" void kernel_launch(void* const* d_in, const int* in_sizes, int n_in,
                              void* d_out, int out_size, void* d_ws, size_t ws_size,
                              hipStream_t stream) {
  const float* X = (const float*)d_in[0];   // (B, G, K) f32
  const float* P = (const float*)d_in[1];   // (G, 16, D) f32
  float* Out     = (float*)d_out;           // (B, G, D) f32

  dim3 grid(Gsz, Bsz / 128);                // 512 x 16 blocks
  dim3 block(256);                          // 8 waves -> 128 b-rows per block
  hipLaunchKernelGGL(hoact_wmma, grid, block, 0, stream, X, P, Out);
}